// BNO_85186381349527
// MI455X (gfx1250) — compile-verified
//
#include <hip/hip_runtime.h>
#include <math.h>

// ---------------------------------------------------------------------------
// Problem constants (from the reference)
// ---------------------------------------------------------------------------
#define BB_   4
#define NX_   8192
#define NY_   2048
#define CC_   128
#define KK_   128
#define JJ_   257          // 2K+1 columns: [cos 0..127 | sin 128..255 | const 256]
#define NL_   4
#define FC_   128
#define SPLITK 8

typedef float v2f __attribute__((ext_vector_type(2)));
typedef float v8f __attribute__((ext_vector_type(8)));

__device__ __forceinline__ v8f wmma4(v2f a, v2f b, v8f c) {
    // V_WMMA_F32_16X16X4_F32 : D(16x16 f32) = A(16x4 f32) * B(4x16 f32) + C
    return __builtin_amdgcn_wmma_f32_16x16x4_f32(false, a, false, b,
                                                 (short)0, c, false, false);
}

__device__ __forceinline__ v8f zero8() {
    v8f z = {0.f,0.f,0.f,0.f,0.f,0.f,0.f,0.f};
    return z;
}

__device__ __forceinline__ float gelu_exact(float v) {
    return 0.5f * v * (1.0f + erff(v * 0.70710678118654752440f));
}

// ---------------------------------------------------------------------------
// Generic batched fp32 WMMA GEMM with optional deterministic split-K:
//   grid.z = batch * splitk ; z -> (batch = z/splitk, ks = z%splitk)
//   each z-block writes its own output slab at C + z*sCb (no atomics).
// Row-major, arbitrary leading dims / batch strides. Optional bias[row],
// accumulate into C, fused exact-GELU epilogue (only when splitk==1).
// Block = 128 threads (4 waves); tile 64x64; K staged in chunks of 16 via LDS.
// ---------------------------------------------------------------------------
#define TM 64
#define TN 64
#define TK 16
#define LSTR 18   // LDS row stride (floats); even -> 8B-aligned ds_read_b64

__global__ __launch_bounds__(128)
void gemm_wmma_f32(const float* __restrict__ A, const float* __restrict__ Bm,
                   float* __restrict__ Cm, const float* __restrict__ bias,
                   int M, int N, int K,
                   long long sAb, long long sBb, long long sCb,
                   int lda, int ldb, int ldc,
                   int accumulate, int fuse_gelu, int splitk)
{
    __shared__ float As[TM][LSTR];   // [row][k]
    __shared__ float Bs[TN][LSTR];   // transposed: [col][k]

    const int bz = blockIdx.z / splitk;
    const int ks = blockIdx.z % splitk;
    A  += (long long)bz * sAb;
    Bm += (long long)bz * sBb;
    Cm += (long long)blockIdx.z * sCb;

    // this block's K-range (TK-aligned chunks)
    int kchunk = K;
    if (splitk > 1) {
        kchunk = (K + splitk - 1) / splitk;
        kchunk = (kchunk + TK - 1) / TK * TK;
    }
    const int kStart = ks * kchunk;
    const int kEnd   = (kStart + kchunk < K) ? (kStart + kchunk) : K;

    const int M0   = blockIdx.y * TM;
    const int N0   = blockIdx.x * TN;
    const int tid  = threadIdx.x;
    const int wave = tid >> 5;
    const int lane = tid & 31;
    const int l16  = lane & 15;
    const int lh   = lane >> 4;      // 0/1: which half of the wave

    v8f acc[4];
    acc[0] = zero8(); acc[1] = zero8(); acc[2] = zero8(); acc[3] = zero8();

    for (int k0 = kStart; k0 < kEnd; k0 += TK) {
        // stage A tile (64 x 16), guarded
        for (int i = tid; i < TM * TK; i += 128) {
            int r = i >> 4, c = i & 15;
            int gr = M0 + r, gk = k0 + c;
            As[r][c] = (gr < M && gk < kEnd) ? A[(long long)gr * lda + gk] : 0.0f;
        }
        // stage B tile (16 x 64) transposed into [n][k], guarded
        for (int i = tid; i < TK * TN; i += 128) {
            int kd = i >> 6, n = i & 63;
            int gk = k0 + kd, gn = N0 + n;
            Bs[n][kd] = (gk < kEnd && gn < N) ? Bm[(long long)gk * ldb + gn] : 0.0f;
        }
        // prefetch next K-chunk into GL2 while this chunk computes
        {
            int nk = k0 + TK;
            if (nk < kEnd) {
                if (tid < TM) {
                    int gr = M0 + tid;
                    if (gr < M)
                        __builtin_prefetch(&A[(long long)gr * lda + nk], 0, 3);
                }
                if (tid >= 64 && tid < 96) {
                    int kd = (tid - 64) >> 1, half = (tid - 64) & 1;
                    int gk = nk + kd, gn = N0 + half * 32;
                    if (gk < kEnd && gn < N)
                        __builtin_prefetch(&Bm[(long long)gk * ldb + gn], 0, 3);
                }
            }
        }
        __syncthreads();

        #pragma unroll
        for (int kc = 0; kc < TK; kc += 4) {
            // A fragment: lane l -> row = wave*16 + (l&15); K = kc + (l>>4)*2 + {0,1}
            v2f a = *(const v2f*)&As[wave * 16 + l16][kc + lh * 2];
            #pragma unroll
            for (int j = 0; j < 4; ++j) {
                v2f b = *(const v2f*)&Bs[j * 16 + l16][kc + lh * 2];
                acc[j] = wmma4(a, b, acc[j]);
            }
        }
        __syncthreads();
    }

    // Epilogue. D layout: VGPR r, lanes 0-15 -> M=r, lanes 16-31 -> M=8+r; N=lane&15
    #pragma unroll
    for (int j = 0; j < 4; ++j) {
        int col = N0 + j * 16 + l16;
        if (col < N) {
            #pragma unroll
            for (int r = 0; r < 8; ++r) {
                int row = M0 + wave * 16 + 8 * lh + r;
                if (row < M) {
                    float v = acc[j][r];
                    if (bias) v += bias[row];
                    long long off = (long long)row * ldc + col;
                    if (accumulate) v += Cm[off];
                    if (fuse_gelu) v = gelu_exact(v);
                    Cm[off] = v;
                }
            }
        }
    }
}

// deterministic fixed-order reduction of split-K partial slabs
__global__ void reduce_splitk(const float* __restrict__ P, float* __restrict__ XC,
                              int splitk)
{
    int t = blockIdx.x * blockDim.x + threadIdx.x;
    const int CJ = CC_ * JJ_;
    if (t >= BB_ * CJ) return;
    int b = t / CJ, r = t % CJ;
    float s = 0.0f;
    for (int i = 0; i < splitk; ++i)
        s += P[(long long)(b * splitk + i) * CJ + r];
    XC[t] = s;
}

// ---------------------------------------------------------------------------
// Per-mode channel mix (WMMA, b padded 4->16):
//   fc[b,o,k] = sum_c xc[b,c,k]*wc[c,o,k] - xs[b,c,k]*ws[c,o,k]
//   fs[b,o,k] = sum_c xs[b,c,k]*wc[c,o,k] + xc[b,c,k]*ws[c,o,k]
//   G[b,o,k] = 2*fc ; G[b,o,128+k] = -2*fs
// XC layout (B, C, J); wc/ws layout (C, C, K); G layout (B, C, J).
// grid = (K, 2), block = 128 (4 waves, 16 o-columns each).
// ---------------------------------------------------------------------------
__global__ __launch_bounds__(128)
void mix_wmma(const float* __restrict__ XC, const float* __restrict__ wc,
              const float* __restrict__ ws, float* __restrict__ G)
{
    const int k    = blockIdx.x;                       // mode index 0..127
    const int o0   = blockIdx.y * 64 + (threadIdx.x >> 5) * 16;
    const int lane = threadIdx.x & 31;
    const int l16  = lane & 15;
    const int lh   = lane >> 4;

    v8f fca = zero8();
    v8f fsa = zero8();

    const int b = l16;                                 // A-matrix row = batch idx
    const int o = o0 + l16;                            // B-matrix col = out chan

    for (int c0 = 0; c0 < CC_; c0 += 4) {
        const int cc = c0 + lh * 2;
        v2f axc, axs;
        if (b < BB_) {
            const float* base = XC + ((long long)b * CC_ + cc) * JJ_;
            axc.x = base[k];            axc.y = base[JJ_ + k];
            axs.x = base[128 + k];      axs.y = base[JJ_ + 128 + k];
        } else {
            axc.x = 0.f; axc.y = 0.f; axs.x = 0.f; axs.y = 0.f;
        }
        long long w0off = (long long)cc * (CC_ * KK_) + (long long)o * KK_ + k;
        v2f bwc, bws, bwsn;
        bwc.x = wc[w0off];             bwc.y = wc[w0off + CC_ * KK_];
        bws.x = ws[w0off];             bws.y = ws[w0off + CC_ * KK_];
        bwsn.x = -bws.x;               bwsn.y = -bws.y;

        fca = wmma4(axc, bwc,  fca);   // + xc*wc
        fca = wmma4(axs, bwsn, fca);   // - xs*ws
        fsa = wmma4(axs, bwc,  fsa);   // + xs*wc
        fsa = wmma4(axc, bws,  fsa);   // + xc*ws
    }

    #pragma unroll
    for (int r = 0; r < 8; ++r) {
        int m = r + 8 * lh;                            // batch index
        if (m < BB_) {
            long long go = ((long long)m * CC_ + o) * JJ_;
            G[go + k]       =  2.0f * fca[r];
            G[go + 128 + k] = -2.0f * fsa[r];
        }
    }
}

// f0[b,o] = sum_c XC[b,c,256] * w0[c,o]  ->  G[b,o,256]
__global__ void f0_kernel(const float* __restrict__ XC,
                          const float* __restrict__ w0, float* __restrict__ G)
{
    const int b = blockIdx.x;
    const int o = threadIdx.x;           // 128 threads
    float s = 0.0f;
    for (int c = 0; c < CC_; ++c)
        s += XC[((long long)b * CC_ + c) * JJ_ + 256] * w0[c * CC_ + o];
    G[((long long)b * CC_ + o) * JJ_ + 256] = s;
}

// ---------------------------------------------------------------------------
// Bases: t = nodes . (modes*sp_L); WB (B,N,J) = weighted [cos|sin|1];
// BBT (B,J,N) = unweighted, transposed. grid=(N,B), block=128 (one thread / k)
// ---------------------------------------------------------------------------
__global__ void bases_kernel(const float* __restrict__ nodes,
                             const float* __restrict__ wts,
                             const float* __restrict__ modes,
                             const float* __restrict__ spL,
                             int N, float* __restrict__ WB,
                             float* __restrict__ BBT)
{
    const int n = blockIdx.x;
    const int b = blockIdx.y;
    const int k = threadIdx.x;           // 0..127
    float m0 = modes[k * 2 + 0] * spL[0];
    float m1 = modes[k * 2 + 1] * spL[1];
    long long bn = (long long)b * N + n;
    float t = nodes[bn * 2 + 0] * m0 + nodes[bn * 2 + 1] * m1;
    float s, c;
    sincosf(t, &s, &c);
    float w = wts[bn];
    long long wb = bn * JJ_;
    WB[wb + k]       = c * w;
    WB[wb + 128 + k] = s * w;
    long long bt = (long long)b * JJ_ * N;
    BBT[bt + (long long)k * N + n]         = c;
    BBT[bt + (long long)(128 + k) * N + n] = s;
    if (k == 0) {
        WB[wb + 256] = w;
        BBT[bt + 256LL * N + n] = 1.0f;
    }
}

// fc0 lift: out[b,c,n] = sum_i w[c,i]*in[b,n,i] + bias[c]
__global__ void fc0_kernel(const float* __restrict__ in, const float* __restrict__ w,
                           const float* __restrict__ bias, float* __restrict__ out,
                           int N, int IN)
{
    long long t = (long long)blockIdx.x * blockDim.x + threadIdx.x;
    long long total = (long long)BB_ * CC_ * N;
    if (t >= total) return;
    int n = (int)(t % N);
    int c = (int)((t / N) % CC_);
    int b = (int)(t / ((long long)CC_ * N));
    const float* ip = in + ((long long)b * N + n) * IN;
    float s = bias[c];
    for (int i = 0; i < IN; ++i) s += w[c * IN + i] * ip[i];
    out[t] = s;                          // t == b*C*N + c*N + n
}

__global__ void gelu_kernel(float* __restrict__ d, long long n)
{
    long long t = (long long)blockIdx.x * blockDim.x + threadIdx.x;
    if (t < n) d[t] = gelu_exact(d[t]);
}

// final: out[b,n] = sum_f fc2_w[f] * Z[b,f,n] + fc2_b
__global__ void fc2_kernel(const float* __restrict__ Z, const float* __restrict__ w,
                           const float* __restrict__ b2, float* __restrict__ out)
{
    int t = blockIdx.x * blockDim.x + threadIdx.x;
    if (t >= BB_ * NX_) return;
    int n = t % NX_;
    int b = t / NX_;
    const float* zp = Z + (long long)b * FC_ * NX_ + n;
    float s = b2[0];
    for (int f = 0; f < FC_; ++f) s += w[f] * zp[(long long)f * NX_];
    out[t] = s;
}

// ---------------------------------------------------------------------------
// Host side
// ---------------------------------------------------------------------------
static inline void launch_gemm(const float* A, const float* B, float* C,
                               const float* bias, int M, int N, int K,
                               long long sA, long long sB, long long sC,
                               int lda, int ldb, int ldc,
                               int acc, int gelu, hipStream_t s)
{
    dim3 grid((N + TN - 1) / TN, (M + TM - 1) / TM, BB_);
    gemm_wmma_f32<<<grid, 128, 0, s>>>(A, B, C, bias, M, N, K,
                                       sA, sB, sC, lda, ldb, ldc, acc, gelu, 1);
}

// split-K projection: XC[b] (C x J) = U[b] (C x Nred) * WB[b] (Nred x J)
static inline void launch_proj_splitk(const float* U, const float* WB, float* PK,
                                      float* XC, int Nred, hipStream_t s)
{
    dim3 grid((JJ_ + TN - 1) / TN, (CC_ + TM - 1) / TM, BB_ * SPLITK);
    gemm_wmma_f32<<<grid, 128, 0, s>>>(U, WB, PK, nullptr, CC_, JJ_, Nred,
                                       (long long)CC_ * Nred,
                                       (long long)Nred * JJ_,
                                       (long long)CC_ * JJ_,
                                       Nred, JJ_, JJ_, 0, 0, SPLITK);
    int total = BB_ * CC_ * JJ_;
    reduce_splitk<<<(total + 255) / 256, 256, 0, s>>>(PK, XC, SPLITK);
}

extern "C" void kernel_launch(void* const* d_in, const int* in_sizes, int n_in,
                              void* d_out, int out_size, void* d_ws, size_t ws_size,
                              hipStream_t stream)
{
    (void)in_sizes; (void)n_in; (void)out_size; (void)ws_size;

    const float* x          = (const float*)d_in[0];
    const float* y          = (const float*)d_in[1];
    /* d_in[2] node_mask_x: unused by reference */
    const float* nodes_x    = (const float*)d_in[3];
    const float* nodes_y    = (const float*)d_in[4];
    const float* nw_x       = (const float*)d_in[5];
    const float* nw_y       = (const float*)d_in[6];
    const float* modes      = (const float*)d_in[7];
    const float* sp_L       = (const float*)d_in[8];
    const float* fc0_x_w    = (const float*)d_in[9];
    const float* fc0_x_b    = (const float*)d_in[10];
    const float* fc0_y_w    = (const float*)d_in[11];
    const float* fc0_y_b    = (const float*)d_in[12];
    const float* ext_wc     = (const float*)d_in[13];
    const float* ext_ws     = (const float*)d_in[14];
    const float* ext_w0     = (const float*)d_in[15];
    const float* spx_wc     = (const float*)d_in[16];
    const float* spx_ws     = (const float*)d_in[17];
    const float* spx_w0     = (const float*)d_in[18];
    const float* spy_wc     = (const float*)d_in[19];
    const float* spy_ws     = (const float*)d_in[20];
    const float* spy_w0     = (const float*)d_in[21];
    const float* wsx_w      = (const float*)d_in[22];
    const float* wsx_b      = (const float*)d_in[23];
    const float* wsy_w      = (const float*)d_in[24];
    const float* wsy_b      = (const float*)d_in[25];
    const float* fc1_w      = (const float*)d_in[26];
    const float* fc1_b      = (const float*)d_in[27];
    const float* fc2_w      = (const float*)d_in[28];
    const float* fc2_b      = (const float*)d_in[29];

    // workspace carve-up (float offsets)
    float* ws = (float*)d_ws;
    long long off = 0;
    float* WB_x  = ws + off; off += (long long)BB_ * NX_ * JJ_;   // 33.7 MB
    float* BBT_x = ws + off; off += (long long)BB_ * JJ_ * NX_;   // 33.7 MB
    float* WB_y  = ws + off; off += (long long)BB_ * NY_ * JJ_;   //  8.4 MB
    float* BBT_y = ws + off; off += (long long)BB_ * JJ_ * NY_;   //  8.4 MB
    float* xA    = ws + off; off += (long long)BB_ * CC_ * NX_;   // 16.8 MB
    float* xB    = ws + off; off += (long long)BB_ * CC_ * NX_;   // 16.8 MB
    float* yA    = ws + off; off += (long long)BB_ * CC_ * NY_;   //  4.2 MB
    float* yB    = ws + off; off += (long long)BB_ * CC_ * NY_;   //  4.2 MB
    float* XCx   = ws + off; off += (long long)BB_ * CC_ * JJ_;
    float* XCy   = ws + off; off += (long long)BB_ * CC_ * JJ_;
    float* G_ext = ws + off; off += (long long)BB_ * CC_ * JJ_;
    float* G_spx = ws + off; off += (long long)BB_ * CC_ * JJ_;
    float* G_spy = ws + off; off += (long long)BB_ * CC_ * JJ_;
    float* PK    = ws + off; off += (long long)BB_ * SPLITK * CC_ * JJ_; // 4.2 MB

    const long long CCK = (long long)CC_ * CC_ * KK_;   // per-layer wc/ws stride
    const long long CCo = (long long)CC_ * CC_;         // per-layer w0 / ws_w stride

    // --- bases ---
    bases_kernel<<<dim3(NX_, BB_), 128, 0, stream>>>(nodes_x, nw_x, modes, sp_L,
                                                     NX_, WB_x, BBT_x);
    bases_kernel<<<dim3(NY_, BB_), 128, 0, stream>>>(nodes_y, nw_y, modes, sp_L,
                                                     NY_, WB_y, BBT_y);
    // --- lifting fc0 ---
    {
        long long tx = (long long)BB_ * CC_ * NX_;
        fc0_kernel<<<(int)((tx + 255) / 256), 256, 0, stream>>>(x, fc0_x_w, fc0_x_b, xA, NX_, 2);
        long long ty = (long long)BB_ * CC_ * NY_;
        fc0_kernel<<<(int)((ty + 255) / 256), 256, 0, stream>>>(y, fc0_y_w, fc0_y_b, yA, NY_, 3);
    }

    float* xCur = xA; float* xNext = xB;
    float* yCur = yA; float* yNext = yB;

    for (int i = 0; i < NL_; ++i) {
        // projection of y (split-K):  XCy[b] = yCur[b] (C x NY) * WB_y[b] (NY x 257)
        launch_proj_splitk(yCur, WB_y, PK, XCy, NY_, stream);
        mix_wmma<<<dim3(KK_, 2), 128, 0, stream>>>(XCy, ext_wc + i * CCK, ext_ws + i * CCK, G_ext);
        f0_kernel<<<BB_, CC_, 0, stream>>>(XCy, ext_w0 + i * CCo, G_ext);

        // projection of x (split-K)
        launch_proj_splitk(xCur, WB_x, PK, XCx, NX_, stream);
        mix_wmma<<<dim3(KK_, 2), 128, 0, stream>>>(XCx, spx_wc + i * CCK, spx_ws + i * CCK, G_spx);
        f0_kernel<<<BB_, CC_, 0, stream>>>(XCx, spx_w0 + i * CCo, G_spx);

        // x3 = wsx_w[i] @ xCur + wsx_b[i]   (overwrite xNext)
        launch_gemm(wsx_w + i * CCo, xCur, xNext, wsx_b + (long long)i * CC_,
                    CC_, NX_, CC_, 0, (long long)CC_ * NX_, (long long)CC_ * NX_,
                    CC_, NX_, NX_, 0, 0, stream);
        // x1 += G_ext @ BBT_x ; x2 += G_spx @ BBT_x
        launch_gemm(G_ext, BBT_x, xNext, nullptr, CC_, NX_, JJ_,
                    (long long)CC_ * JJ_, (long long)JJ_ * NX_, (long long)CC_ * NX_,
                    JJ_, NX_, NX_, 1, 0, stream);
        launch_gemm(G_spx, BBT_x, xNext, nullptr, CC_, NX_, JJ_,
                    (long long)CC_ * JJ_, (long long)JJ_ * NX_, (long long)CC_ * NX_,
                    JJ_, NX_, NX_, 1, 0, stream);

        if (i < NL_ - 1) {
            long long nx = (long long)BB_ * CC_ * NX_;
            gelu_kernel<<<(int)((nx + 255) / 256), 256, 0, stream>>>(xNext, nx);

            // y update
            mix_wmma<<<dim3(KK_, 2), 128, 0, stream>>>(XCy, spy_wc + i * CCK, spy_ws + i * CCK, G_spy);
            f0_kernel<<<BB_, CC_, 0, stream>>>(XCy, spy_w0 + i * CCo, G_spy);
            launch_gemm(wsy_w + i * CCo, yCur, yNext, wsy_b + (long long)i * CC_,
                        CC_, NY_, CC_, 0, (long long)CC_ * NY_, (long long)CC_ * NY_,
                        CC_, NY_, NY_, 0, 0, stream);
            launch_gemm(G_spy, BBT_y, yNext, nullptr, CC_, NY_, JJ_,
                        (long long)CC_ * JJ_, (long long)JJ_ * NY_, (long long)CC_ * NY_,
                        JJ_, NY_, NY_, 1, 0, stream);
            long long ny = (long long)BB_ * CC_ * NY_;
            gelu_kernel<<<(int)((ny + 255) / 256), 256, 0, stream>>>(yNext, ny);
            { float* t = yCur; yCur = yNext; yNext = t; }
        }
        { float* t = xCur; xCur = xNext; xNext = t; }
    }

    // fc1 (with fused exact GELU) into the free ping buffer
    float* Z = xNext;
    launch_gemm(fc1_w, xCur, Z, fc1_b, FC_, NX_, CC_,
                0, (long long)CC_ * NX_, (long long)FC_ * NX_,
                CC_, NX_, NX_, 0, 1, stream);
    // fc2 -> d_out (B*NX floats)
    fc2_kernel<<<(BB_ * NX_ + 255) / 256, 256, 0, stream>>>(Z, fc2_w, fc2_b, (float*)d_out);
}